// SSDLoss_30485677867331
// MI455X (gfx1250) — compile-verified
//
#include <hip/hip_runtime.h>
#include <cstdint>
#include <cstddef>

#define BB 32
#define DD 24564
#define AA 50
#define NCLS 81      // C+1
#define NEGR 3
#define MTHR 0.5f
#define CHUNKF 6912  // floats per LDS chunk; 3 * 6912 = 256 * 81 exactly

typedef __attribute__((ext_vector_type(2))) float v2f;
typedef __attribute__((ext_vector_type(8))) float v8f;

// Wave32 sum reduction on the matrix pipe.
// A(16x4): lane L<16 -> A[L][0]=x; lane L>=16 -> A[L-16][2]=x; other K slots 0.
// B(4x16) = ones  =>  D[i][j] = p_i + p_{i+16}.
// Sum of the 8 D vgprs leaves half-wave totals; one xor-16 shuffle finishes.
__device__ __forceinline__ float wave_sum_wmma(float x) {
  v2f a; a[0] = x;    a[1] = 0.0f;
  v2f b; b[0] = 1.0f; b[1] = 1.0f;
  v8f c = {0.f,0.f,0.f,0.f,0.f,0.f,0.f,0.f};
  v8f d = __builtin_amdgcn_wmma_f32_16x16x4_f32(false, a, false, b, (short)0, c, false, false);
  float s = ((d[0]+d[1])+(d[2]+d[3])) + ((d[4]+d[5])+(d[6]+d[7]));
  s += __shfl_xor(s, 16, 32);
  return s;
}

// CDNA5 async DMA: 16B global -> LDS, tracked by ASYNCcnt (no VGPR round-trip).
__device__ __forceinline__ void async_copy_b128(uint32_t ldsoff, uint64_t gaddr) {
  asm volatile("global_load_async_to_lds_b128 %0, %1, off"
               :: "v"(ldsoff), "v"(gaddr) : "memory");
}

__global__ void k_init(int* Ncnt, float* pos_sum, float* reg_sum, float* acc) {
  int t = threadIdx.x;
  if (t == 0) *Ncnt = 0;
  if (t < BB) { pos_sum[t] = 0.f; reg_sum[t] = 0.f; }
  if (t < 2)  acc[t] = 0.f;
}

// Per-box: best annotation (first-index argmax over A) and matched = maxIOU >= 0.5
__global__ void k_match(const float* __restrict__ def, const float* __restrict__ tgt,
                        int* __restrict__ matched, int* __restrict__ bwa) {
  __shared__ float ax0[AA], ay0[AA], ax1[AA], ay1[AA], aarea[AA];
  int tid = threadIdx.x;
  if (tid < AA) {
    float cx = tgt[tid*6+2], cy = tgt[tid*6+3], w = tgt[tid*6+4], h = tgt[tid*6+5];
    float x0 = fmaxf(cx - 0.5f*w, 0.f), y0 = fmaxf(cy - 0.5f*h, 0.f);
    float x1 = fminf(cx + 0.5f*w, 1.f), y1 = fminf(cy + 0.5f*h, 1.f);
    ax0[tid]=x0; ay0[tid]=y0; ax1[tid]=x1; ay1[tid]=y1; aarea[tid]=(x1-x0)*(y1-y0);
  }
  __syncthreads();
  int d = blockIdx.x*blockDim.x + tid;
  if (d >= DD) return;
  float cx = def[d*4+0], cy = def[d*4+1], w = def[d*4+2], h = def[d*4+3];
  float dx0 = fmaxf(cx-0.5f*w, 0.f), dy0 = fmaxf(cy-0.5f*h, 0.f);
  float dx1 = fminf(cx+0.5f*w, 1.f), dy1 = fminf(cy+0.5f*h, 1.f);
  float darea = (dx1-dx0)*(dy1-dy0);
  float best = -1.f; int bi = 0;
  for (int a = 0; a < AA; ++a) {
    float ix0 = fmaxf(ax0[a], dx0), iy0 = fmaxf(ay0[a], dy0);
    float ix1 = fminf(ax1[a], dx1), iy1 = fminf(ay1[a], dy1);
    float iw = fmaxf(ix1-ix0, 0.f), ih = fmaxf(iy1-iy0, 0.f);
    float inter = iw*ih;
    float iou = inter / (aarea[a] + darea - inter);
    if (iou > best) { best = iou; bi = a; }   // strict > => first index on ties
  }
  matched[d] = (best >= MTHR) ? 1 : 0;
  bwa[d] = bi;
}

// Per-annotation: first-index argmax over the D boxes.
__global__ void k_ann_argmax(const float* __restrict__ def, const float* __restrict__ tgt,
                             int* __restrict__ awb) {
  __shared__ float sv[256];
  __shared__ int   si[256];
  int a = blockIdx.x, tid = threadIdx.x;
  float cx = tgt[a*6+2], cy = tgt[a*6+3], w = tgt[a*6+4], h = tgt[a*6+5];
  float ax0 = fmaxf(cx - 0.5f*w, 0.f), ay0 = fmaxf(cy - 0.5f*h, 0.f);
  float ax1 = fminf(cx + 0.5f*w, 1.f), ay1 = fminf(cy + 0.5f*h, 1.f);
  float aarea = (ax1-ax0)*(ay1-ay0);
  float best = -1.f; int bi = 0x7fffffff;
  for (int d = tid; d < DD; d += blockDim.x) {
    float c0 = def[d*4+0], c1 = def[d*4+1], c2 = def[d*4+2], c3 = def[d*4+3];
    float dx0 = fmaxf(c0-0.5f*c2, 0.f), dy0 = fmaxf(c1-0.5f*c3, 0.f);
    float dx1 = fminf(c0+0.5f*c2, 1.f), dy1 = fminf(c1+0.5f*c3, 1.f);
    float darea = (dx1-dx0)*(dy1-dy0);
    float ix0 = fmaxf(ax0, dx0), iy0 = fmaxf(ay0, dy0);
    float ix1 = fminf(ax1, dx1), iy1 = fminf(ay1, dy1);
    float iw = fmaxf(ix1-ix0, 0.f), ih = fmaxf(iy1-iy0, 0.f);
    float inter = iw*ih;
    float iou = inter / (aarea + darea - inter);
    if (iou > best || (iou == best && d < bi)) { best = iou; bi = d; }
  }
  sv[tid] = best; si[tid] = bi; __syncthreads();
  for (int off = blockDim.x>>1; off; off >>= 1) {
    if (tid < off) {
      if (sv[tid+off] > sv[tid] || (sv[tid+off] == sv[tid] && si[tid+off] < si[tid])) {
        sv[tid] = sv[tid+off]; si[tid] = si[tid+off];
      }
    }
    __syncthreads();
  }
  if (tid == 0) awb[a] = si[0];
}

__global__ void k_scatter(const int* __restrict__ awb, int* __restrict__ matched,
                          int* __restrict__ bwa) {
  int a = threadIdx.x;
  if (a < AA) { int j = awb[a]; matched[j] = 1; bwa[j] = a; }
}

// true_cls, true_offsets, N = sum(matched)
__global__ void k_assign(const float* __restrict__ def, const float* __restrict__ tgt,
                         const int* __restrict__ matched, const int* __restrict__ bwa,
                         int* __restrict__ tcls, float* __restrict__ toff,
                         int* __restrict__ Ncnt) {
  int d = blockIdx.x*blockDim.x + threadIdx.x;
  if (d >= DD) return;
  int m = matched[d], a = bwa[d];
  tcls[d] = m ? (int)tgt[a*6+1] : 0;
  float mcx=tgt[a*6+2], mcy=tgt[a*6+3], mw=tgt[a*6+4], mh=tgt[a*6+5];
  float dcx=def[d*4+0], dcy=def[d*4+1], dw=def[d*4+2], dh=def[d*4+3];
  toff[d*4+0] = (mcx-dcx)/(dw*0.1f);
  toff[d*4+1] = (mcy-dcy)/(dh*0.1f);
  toff[d*4+2] = __logf(mw/dw)*5.0f;   // /0.2
  toff[d*4+3] = __logf(mh/dh)*5.0f;
  if (m) atomicAdd(Ncnt, 1);
}

// Streaming pass. The 256-box tile of predicted_classes is contiguous in global
// (classes innermost), so it is DMA'd into LDS in three 27,648B chunks with
// global_load_async_to_lds_b128 (coalesced 16B transfers, ASYNCcnt-tracked),
// then consumed bank-conflict-free (stride 81 -> 17 mod 64, gcd 1).
// Online logsumexp keeps the pass single-read. WMMA wave reductions at the end
// (EXEC all-ones: out-of-range lanes contribute zero, never early-return).
__global__ void k_ce(const float* __restrict__ pcls, const float* __restrict__ poff,
                     const int* __restrict__ matched, const int* __restrict__ tcls,
                     const float* __restrict__ toff, float* __restrict__ neg_ce,
                     float* __restrict__ pos_sum, float* __restrict__ reg_sum) {
  __shared__ __attribute__((aligned(16))) float smem[CHUNKF];
  int tid = threadIdx.x;
  int b = blockIdx.y;
  int d0 = blockIdx.x * 256;
  int d = d0 + tid;
  bool valid = d < DD;
  int dc = valid ? d : (DD-1);
  int nbox = min(DD - d0, 256);
  int tile_floats = nbox * NCLS;
  const float* gtile = pcls + ((size_t)b*DD + d0) * NCLS;  // 16B aligned
  uint32_t lds_base = (uint32_t)(uintptr_t)(&smem[0]);     // low 32b of flat = LDS offset
  int tc = tcls[dc];
  int mylo = tid * NCLS;
  float mx = -__builtin_inff(), s = 0.f, xt = 0.f;

  for (int j = 0; j < 3; ++j) {
    int cbase = j * CHUNKF;
    int cnt = min(CHUNKF, tile_floats - cbase);  // always multiple of 4 here
    if (cnt > 0) {
      int nf4 = cnt >> 2;
      for (int f = tid; f < nf4; f += 256) {
        uint64_t ga = (uint64_t)(uintptr_t)(gtile + cbase + f*4);
        async_copy_b128(lds_base + (uint32_t)(f*16), ga);
      }
      asm volatile("s_wait_asynccnt 0x0" ::: "memory");
    }
    __syncthreads();
    int lo = (mylo > cbase) ? mylo : cbase;
    int hi = min(mylo + NCLS, cbase + cnt);
    if (!valid) hi = lo;
    for (int f = lo; f < hi; ++f) {
      float x = smem[f - cbase];
      int c = f - mylo;
      if (c == tc) xt = x;
      if (x <= mx) { s += __expf(x - mx); }
      else         { s = s * __expf(mx - x) + 1.0f; mx = x; }
    }
    __syncthreads();
  }

  float ce = (mx + __logf(s)) - xt;
  int mt = matched[dc];
  float posv = (valid && mt) ? ce : 0.f;
  float regv = 0.f;
  if (valid && mt) {
    const float* po = poff + ((size_t)b*DD + dc)*4;
    const float* to = toff + (size_t)dc*4;
    #pragma unroll
    for (int k = 0; k < 4; ++k) {
      float diff = fabsf(po[k] - to[k]);
      regv += (diff < 1.f) ? 0.5f*diff*diff : (diff - 0.5f);
    }
  }
  if (valid) neg_ce[(size_t)b*DD + d] = mt ? -__builtin_inff() : ce;

  float ps = wave_sum_wmma(posv);
  float rs = wave_sum_wmma(regv);
  if ((threadIdx.x & 31) == 0) {
    if (ps != 0.f) atomicAdd(&pos_sum[b], ps);
    if (rs != 0.f) atomicAdd(&reg_sum[b], rs);
  }
}

// Per-image top-3N sum via exact bit-pattern binary search (CE >= 0 so uint order == float order).
// The per-image 96KB CE slab is L2-resident (192MB L2), so the counting passes are cheap.
__global__ void k_topk(const float* __restrict__ neg_ce, const int* __restrict__ Ncnt,
                       const float* __restrict__ pos_sum, const float* __restrict__ reg_sum,
                       float* __restrict__ acc) {
  __shared__ int   sci[256];
  __shared__ float scf[256];
  int b = blockIdx.x, tid = threadIdx.x;
  int N = *Ncnt;
  float Nf = (float)N;
  int k_eff = min(NEGR * N, DD - N);   // finite entries = D - N
  const float* neg = neg_ce + (size_t)b*DD;
  unsigned lo = 0u, hi = 0x7f000000u;
  while (lo < hi) {
    unsigned mid = lo + ((hi - lo + 1u) >> 1);
    float t = __uint_as_float(mid);
    int cnt = 0;
    for (int d = tid; d < DD; d += blockDim.x) cnt += (neg[d] >= t) ? 1 : 0;
    sci[tid] = cnt; __syncthreads();
    for (int off = blockDim.x>>1; off; off >>= 1) {
      if (tid < off) sci[tid] += sci[tid+off];
      __syncthreads();
    }
    int total = sci[0]; __syncthreads();
    if (total >= k_eff) lo = mid; else hi = mid - 1u;
  }
  float v = __uint_as_float(lo);   // k_eff-th largest value
  float ssum = 0.f; int cgt = 0;
  for (int d = tid; d < DD; d += blockDim.x) {
    float x = neg[d];
    if (x > v) { ssum += x; cgt++; }
  }
  scf[tid] = ssum; sci[tid] = cgt; __syncthreads();
  for (int off = blockDim.x>>1; off; off >>= 1) {
    if (tid < off) { scf[tid] += scf[tid+off]; sci[tid] += sci[tid+off]; }
    __syncthreads();
  }
  if (tid == 0) {
    float negsum = (k_eff > 0) ? (scf[0] + (float)(k_eff - sci[0]) * v) : 0.f;
    float cls_l = (pos_sum[b] + negsum) / Nf;
    float reg_l = reg_sum[b] / Nf;
    atomicAdd(&acc[0], cls_l);
    atomicAdd(&acc[1], reg_l);
  }
}

__global__ void k_final(const float* __restrict__ acc, float* __restrict__ out) {
  if (threadIdx.x == 0) {
    float cls = acc[0] / (float)BB;
    float loc = acc[1] / (float)BB;
    out[0] = cls + loc;   // total
    out[1] = loc;         // localization
    out[2] = cls;         // classification
  }
}

extern "C" void kernel_launch(void* const* d_in, const int* in_sizes, int n_in,
                              void* d_out, int out_size, void* d_ws, size_t ws_size,
                              hipStream_t stream) {
  const float* poff = (const float*)d_in[0];   // (B, D, 4)
  const float* pcls = (const float*)d_in[1];   // (B, D, 81)
  const float* tgt  = (const float*)d_in[2];   // (B*A, 6)
  const float* def  = (const float*)d_in[3];   // (D, 4)

  char* ws = (char*)d_ws;
  size_t off = 0;
  auto alloc = [&](size_t bytes) -> void* {
    void* p = ws + off;
    off = (off + bytes + 255) & ~(size_t)255;
    return p;
  };
  float* neg_ce  = (float*)alloc(sizeof(float)*(size_t)BB*DD);
  int*   matched = (int*)  alloc(sizeof(int)*DD);
  int*   bwa     = (int*)  alloc(sizeof(int)*DD);
  int*   awb     = (int*)  alloc(sizeof(int)*64);
  int*   tcls    = (int*)  alloc(sizeof(int)*DD);
  float* toff    = (float*)alloc(sizeof(float)*4*DD);
  int*   Ncnt    = (int*)  alloc(256);
  float* pos_sum = (float*)alloc(sizeof(float)*BB);
  float* reg_sum = (float*)alloc(sizeof(float)*BB);
  float* acc     = (float*)alloc(sizeof(float)*2);

  int nblk = (DD + 255) / 256;
  k_init      <<<1, 128, 0, stream>>>(Ncnt, pos_sum, reg_sum, acc);
  k_match     <<<nblk, 256, 0, stream>>>(def, tgt, matched, bwa);
  k_ann_argmax<<<AA, 256, 0, stream>>>(def, tgt, awb);
  k_scatter   <<<1, 64, 0, stream>>>(awb, matched, bwa);
  k_assign    <<<nblk, 256, 0, stream>>>(def, tgt, matched, bwa, tcls, toff, Ncnt);
  dim3 g5(nblk, BB);
  k_ce        <<<g5, 256, 0, stream>>>(pcls, poff, matched, tcls, toff,
                                       neg_ce, pos_sum, reg_sum);
  k_topk      <<<BB, 256, 0, stream>>>(neg_ce, Ncnt, pos_sum, reg_sum, acc);
  k_final     <<<1, 32, 0, stream>>>(acc, (float*)d_out);
}